// Net_79018808312147
// MI455X (gfx1250) — compile-verified
//
#include <hip/hip_runtime.h>

#define DW 4096
#define TILE_M 16
#define NWAVES 32
#define KPW (DW / NWAVES)   // 128 K-elements per wave

typedef __attribute__((ext_vector_type(2))) float v2f;
typedef __attribute__((ext_vector_type(4))) float v4f;
typedef __attribute__((ext_vector_type(8))) float v8f;

// One block = one 16-row output tile, 32 waves split the K dimension.
// Dot-product engine: V_WMMA_F32_16X16X4_F32 with B broadcast across all 16
// columns, so every column of the 16x16 accumulator equals the dot product.
// K-slot remap for wide loads: over a pair of WMMAs covering 8 global k's,
// lanes 0-15 carry {k0..k0+3} and lanes 16-31 carry {k0+4..k0+7}; each lane
// then needs exactly one contiguous 16-byte (b128) load per 8-k step.
__global__ __launch_bounds__(1024) void masked_gemv_wmma(
    const float* __restrict__ W,
    const float* __restrict__ O,      // nullptr -> unmasked (input layer)
    const float* __restrict__ bias,
    const float* __restrict__ src,    // node buffer / x
    const int*   __restrict__ gidx,   // nullptr -> identity gather
    const int*   __restrict__ sidx,   // nullptr -> write pre-scatter output
    float*       __restrict__ dst)
{
    __shared__ float inp_s[DW];
    __shared__ float red[NWAVES][TILE_M];

    const int tid = threadIdx.x;

    // cooperative gather of the full input vector into LDS (16 KB)
    if (gidx) {
        for (int k = tid; k < DW; k += 1024) inp_s[k] = src[gidx[k]];
    } else {
        for (int k = tid; k < DW; k += 1024) inp_s[k] = src[k];
    }
    __syncthreads();

    const int wave = tid >> 5;
    const int lane = tid & 31;
    const int m    = lane & 15;
    const int k4   = (lane >> 4) << 2;      // 0 or 4: half-wave K base
    const int rb   = blockIdx.x * TILE_M;
    const int kbeg = wave * KPW;

    const float* wrow = W + (size_t)(rb + m) * DW;

    v8f acc = {0.f, 0.f, 0.f, 0.f, 0.f, 0.f, 0.f, 0.f};

    if (O) {  // uniform branch: EXEC stays all-ones around WMMA
        const float* orow = O + (size_t)(rb + m) * DW;
        #pragma unroll 4
        for (int k0 = kbeg; k0 < kbeg + KPW; k0 += 8) {
            v4f w = __builtin_nontemporal_load((const v4f*)(wrow + k0 + k4));
            v4f o = __builtin_nontemporal_load((const v4f*)(orow + k0 + k4));
            w *= o;                                         // W .* order mask
            v4f iv = *(const v4f*)(inp_s + k0 + k4);
            v2f a0 = {w.x, w.y},  a1 = {w.z, w.w};
            v2f b0 = {iv.x, iv.y}, b1 = {iv.z, iv.w};
            acc = __builtin_amdgcn_wmma_f32_16x16x4_f32(
                false, a0, false, b0, (short)0, acc, false, false);
            acc = __builtin_amdgcn_wmma_f32_16x16x4_f32(
                false, a1, false, b1, (short)0, acc, false, false);
        }
    } else {
        #pragma unroll 4
        for (int k0 = kbeg; k0 < kbeg + KPW; k0 += 8) {
            v4f w = __builtin_nontemporal_load((const v4f*)(wrow + k0 + k4));
            v4f iv = *(const v4f*)(inp_s + k0 + k4);
            v2f a0 = {w.x, w.y},  a1 = {w.z, w.w};
            v2f b0 = {iv.x, iv.y}, b1 = {iv.z, iv.w};
            acc = __builtin_amdgcn_wmma_f32_16x16x4_f32(
                false, a0, false, b0, (short)0, acc, false, false);
            acc = __builtin_amdgcn_wmma_f32_16x16x4_f32(
                false, a1, false, b1, (short)0, acc, false, false);
        }
    }

    // lanes 0 / 16 hold column N==0: M = 0..7 / 8..15 in acc[0..7]
    if ((lane & 15) == 0) {
        const int mh = (lane >> 4) * 8;
        #pragma unroll
        for (int r = 0; r < 8; ++r) red[wave][mh + r] = acc[r];
    }
    __syncthreads();

    // cross-wave K reduction + bias + relu + scatter
    if (tid < TILE_M) {
        float s = 0.f;
        #pragma unroll
        for (int w = 0; w < NWAVES; ++w) s += red[w][tid];
        s += bias[rb + tid];
        s = fmaxf(s, 0.f);
        const int row = rb + tid;
        if (sidx) dst[sidx[row]] = s;   // permutation scatter into node buffer
        else      dst[row]       = s;   // final layer: pre-scatter output
    }
}

extern "C" void kernel_launch(void* const* d_in, const int* in_sizes, int n_in,
                              void* d_out, int out_size, void* d_ws, size_t ws_size,
                              hipStream_t stream)
{
    const float* x       = (const float*)d_in[0];
    const float* W_in    = (const float*)d_in[1];
    const float* b_in    = (const float*)d_in[2];
    const float* weights = (const float*)d_in[3];
    const float* orders  = (const float*)d_in[4];
    const float* biases  = (const float*)d_in[5];
    const int*   gidx    = (const int*)d_in[6];   // [4][4096]
    const int*   sidx    = (const int*)d_in[7];   // [5][4096]
    float* out  = (float*)d_out;

    float* bufA = (float*)d_ws;        // node buffer ping
    float* bufB = bufA + DW;           // node buffer pong (32 KB total)

    const size_t WSZ = (size_t)DW * DW;
    dim3 grid(DW / TILE_M);            // 256 row tiles
    dim3 block(1024);                  // 32 waves

    // input layer: relu(W_in @ x + b_in), scattered via scatter_idx[0]
    masked_gemv_wmma<<<grid, block, 0, stream>>>(
        W_in, nullptr, b_in, x, nullptr, sidx + 0 * DW, bufA);
    // masked layers 0..2: gather -> (W.*O)@inp -> relu -> scatter
    masked_gemv_wmma<<<grid, block, 0, stream>>>(
        weights + 0 * WSZ, orders + 0 * WSZ, biases + 0 * DW,
        bufA, gidx + 0 * DW, sidx + 1 * DW, bufB);
    masked_gemv_wmma<<<grid, block, 0, stream>>>(
        weights + 1 * WSZ, orders + 1 * WSZ, biases + 1 * DW,
        bufB, gidx + 1 * DW, sidx + 2 * DW, bufA);
    masked_gemv_wmma<<<grid, block, 0, stream>>>(
        weights + 2 * WSZ, orders + 2 * WSZ, biases + 2 * DW,
        bufA, gidx + 2 * DW, sidx + 3 * DW, bufB);
    // last layer: result is the PRE-scatter output -> d_out directly
    masked_gemv_wmma<<<grid, block, 0, stream>>>(
        weights + 3 * WSZ, orders + 3 * WSZ, biases + 3 * DW,
        bufB, gidx + 3 * DW, nullptr, out);
}